// AnalyticHashDepthwiseConv2d_64192581206211
// MI455X (gfx1250) — compile-verified
//
#include <hip/hip_runtime.h>

// Depthwise 7x7 conv, fp32, via V_WMMA_F32_16X16X4_F32 banded-Toeplitz matmul.
// out[n,c,h,w] = sum_{kh,kw} x[n,c,h+kh-3,w+kw-3] * wgt[c,kh,kw] + bias[c]
// wgt[c,kh,kw] = codebook[(c*1337 + kh*7919 + kw*2971 + 3*104729) % 4096]
//               * ((c*4099 + kh*6151 + kw*14887 + 3*4128) % 2 ? +1 : -1)
//
// Per fixed kh:  out_tile(16x16) += A(16x24 input slab) x B(24x16)
//   A[h][k] = xpad[h0+h+kh][w0+k]          (xpad = x with 3-pixel zero border)
//   B[k][w] = filt[kh][k-w] for 0<=k-w<=6  (position-independent Toeplitz band)
// K=24 -> 6 WMMAs of K=4 per kh, 42 per tile.  B fragments are per-channel
// constants: computed ONCE per wave into VGPRs, reused across all 6 h-tiles.

typedef __attribute__((ext_vector_type(2))) float v2f;
typedef __attribute__((ext_vector_type(8))) float v8f;

#define NB      16
#define CH      512
#define HH      96
#define WW      96
#define LDS_ROWS  102     // 96 rows + 6 halo (rows -3..98)
#define LDS_COLS  104     // cols -3..100
#define LDS_PITCH 106     // even (b64-aligned); 106%64=42 -> at most 2-way conflict
#define NWAVES  6         // one wave per 16-wide column tile across W=96
#define NTHREADS (NWAVES * 32)

__global__ __launch_bounds__(NTHREADS) void dwconv_hash_wmma(
    const float* __restrict__ x, const float* __restrict__ codebook,
    const float* __restrict__ bias, float* __restrict__ out)
{
    __shared__ __align__(16) float tile[LDS_ROWS * LDS_PITCH];  // ~43.2 KB
    __shared__ float filt[56];   // 7x7 filter for this channel

    const int bid = blockIdx.x;            // one block per (n, c) plane
    const int c   = bid % CH;
    const int n   = bid / CH;

    const int tid  = threadIdx.x;
    const int lane = tid & 31;
    const int wid  = tid >> 5;

    // ---- per-channel analytic filter -> LDS (49 taps) ----
    if (tid < 49) {
        const int kh = tid / 7, kw = tid % 7;
        const int idx = (c * 1337 + kh * 7919 + kw * 2971 + 3 * 104729) % 4096;
        const int bit = (c * 4099 + kh * 6151 + kw * 14887 + 3 * (4099 + 29)) & 1;
        const float sgn = bit ? 1.0f : -1.0f;
        filt[tid] = codebook[idx] * sgn;
    }

    // ---- stage whole zero-padded plane into LDS (read exactly once) ----
    const float* xplane = x + (size_t)(n * CH + c) * (HH * WW);
    for (int i = tid; i < LDS_ROWS * LDS_COLS; i += NTHREADS) {
        const int r   = i / LDS_COLS;
        const int col = i - r * LDS_COLS;
        const int gr  = r - 3;
        const int gc  = col - 3;
        float v = 0.0f;
        if ((unsigned)gr < (unsigned)HH && (unsigned)gc < (unsigned)WW)
            v = xplane[gr * WW + gc];
        tile[r * LDS_PITCH + col] = v;
    }
    __syncthreads();

    // ---- precompute all 42 B fragments (per-channel Toeplitz) into VGPRs ----
    const int w0 = wid * 16;
    const int m  = lane & 15;          // A: row M ; B: col N
    const int ko = (lane >> 4) * 2;    // half-wave K offset (K=0,1 vs K=2,3)

    v2f bfrag[7][6];
#pragma unroll
    for (int kh = 0; kh < 7; ++kh) {
#pragma unroll
        for (int s = 0; s < 6; ++s) {
            const int kbase = s * 4 + ko;
            const int kw0 = kbase - m;         // B[k][w] band index k-w
            const int kw1 = kbase + 1 - m;
            const int ci0 = kw0 < 0 ? 0 : (kw0 > 6 ? 6 : kw0);
            const int ci1 = kw1 < 0 ? 0 : (kw1 > 6 ? 6 : kw1);
            const float f0 = filt[kh * 7 + ci0];
            const float f1 = filt[kh * 7 + ci1];
            v2f b;
            b[0] = ((unsigned)kw0 <= 6u) ? f0 : 0.0f;
            b[1] = ((unsigned)kw1 <= 6u) ? f1 : 0.0f;
            bfrag[kh][s] = b;
        }
    }

    const float bv = bias[c];
    float* oplane = out + (size_t)(n * CH + c) * (HH * WW);
    const int ncol  = w0 + m;
    const int mbase = (lane >> 4) * 8;

    // ---- 6 h-tiles per wave, reusing the B fragments ----
#pragma unroll 1
    for (int th = 0; th < HH / 16; ++th) {
        const int h0 = th * 16;

        v8f acc = {};
#pragma unroll
        for (int kh = 0; kh < 7; ++kh) {
#pragma unroll
            for (int s = 0; s < 6; ++s) {
                const int kbase = s * 4 + ko;
                // A fragment: slab row = h0 + M + kh, slab col = w0 + k
                const v2f a = *(const v2f*)
                    &tile[(h0 + m + kh) * LDS_PITCH + (w0 + kbase)];
                // D = A(16x4 f32) * B(4x16 f32) + C
                acc = __builtin_amdgcn_wmma_f32_16x16x4_f32(
                    false, a, false, bfrag[kh][s], (short)0, acc, false, false);
            }
        }

        // C/D layout: VGPR i -> M = i + 8*(lane>=16), N = lane%16
#pragma unroll
        for (int i = 0; i < 8; ++i) {
            oplane[(h0 + mbase + i) * WW + ncol] = acc[i] + bv;
        }
    }
}

extern "C" void kernel_launch(void* const* d_in, const int* in_sizes, int n_in,
                              void* d_out, int out_size, void* d_ws, size_t ws_size,
                              hipStream_t stream) {
    const float* x        = (const float*)d_in[0];
    const float* codebook = (const float*)d_in[1];
    const float* bias     = (const float*)d_in[2];
    float* out            = (float*)d_out;

    const int nblocks = NB * CH;   // 16*512 = 8192 blocks, one (n,c) plane each
    dwconv_hash_wmma<<<nblocks, NTHREADS, 0, stream>>>(x, codebook, bias, out);
}